// CC2TBAELoss_23785528885470
// MI455X (gfx1250) — compile-verified
//
#include <hip/hip_runtime.h>

// Problem constants from the reference: N=20000, d=8, D=32
#define NN   20000
#define DD   32
#define dd   8
#define WPB  8                 // waves per block (wave32)
#define BLK  (WPB * 32)        // 256 threads
#define NBLK ((NN + WPB - 1) / WPB)   // 2500, exact

typedef __attribute__((ext_vector_type(2))) float v2f;
typedef __attribute__((ext_vector_type(4))) float v4f;
typedef __attribute__((ext_vector_type(8))) float v8f;

// ---------------------------------------------------------------------------
// Main fused kernel: one wave32 per sample n.
//   Phase A: stage dpi in LDS zero-padded to 16x32, accumulate sum(dpi^2)
//   Phase B: T = pad16(dpi) @ cov        via v_wmma_f32_16x16x4_f32 (16 ops)
//   Phase C: bbt = T @ pad(dpi)^T        via v_wmma_f32_16x16x4_f32 (8 ops)
//   Phase D: qv from decoder_hessian, t = drift - 0.5*qv, mse term
//   Phase E: npv = t - P@t, curvature + tangent(diff^2) terms
//   Phase F: stream encoder_hessian^2 (non-temporal b128 loads)
//   Reduce wave -> block partial -> d_ws[blockIdx.x]
// ---------------------------------------------------------------------------
__global__ __launch_bounds__(BLK) void cc2_loss_main(
    const float* __restrict__ x_hat,
    const float* __restrict__ dpi,
    const float* __restrict__ mp,     // model_projection
    const float* __restrict__ dh,     // decoder_hessian (N,32,8,8)
    const float* __restrict__ eh,     // encoder_hessian (N,8,32,32)
    const float* __restrict__ x,
    const float* __restrict__ drift,
    const float* __restrict__ cov,    // ambient_cov (N,32,32)
    const float* __restrict__ op,     // observed_projection (N,32,32)
    float* __restrict__ partial)
{
    const int lane = threadIdx.x & 31;
    const int wave = threadIdx.x >> 5;
    const int n    = blockIdx.x * WPB + wave;   // always < NN (2500*8 == 20000)

    __shared__ float s_dpi[WPB][16 * DD];   // dpi zero-padded to 16x32
    __shared__ float s_T  [WPB][16 * DD];   // 16x32 T = pad(dpi) @ cov
    __shared__ float s_bbt[WPB][dd * dd];   // 8x8 bbt
    __shared__ float s_t  [WPB][DD];        // tangent vector
    __shared__ float s_part[WPB];

    float sum_mse = 0.f, sum_contr = 0.f, sum_hess = 0.f,
          sum_tang = 0.f, sum_curv = 0.f;

    const int half = lane >> 4;     // which 16-lane half of the wave
    const int lm   = lane & 15;

    // ---- Phase A: stage dpi zero-padded (coalesced: lane = column) ---------
    {
        const float* dp = dpi + (size_t)n * (dd * DD);
        #pragma unroll
        for (int k = 0; k < dd; ++k) {
            float v = dp[k * DD + lane];
            s_dpi[wave][k * DD + lane] = v;
            sum_contr += v * v;
        }
        #pragma unroll
        for (int k = dd; k < 16; ++k)
            s_dpi[wave][k * DD + lane] = 0.f;   // padding rows 8..15
    }
    __syncthreads();

    // ---- Phase B: T(16x32) = pad16(dpi) @ cov, two 16x16 tiles, K=32 in 8 steps
    // ISA layouts (wave32): A 16x4 f32: VGPR v, half h -> K = v + 2h, M = lm
    //                       B 4x16 f32: VGPR v, half h -> K = v + 2h, N = lm
    //                       C/D 16x16 : VGPR v, half h -> M = v + 8h, N = lm
    v8f accT0 = {}; v8f accT1 = {};
    {
        const float* cv = cov + (size_t)n * (DD * DD);
        #pragma unroll
        for (int s = 0; s < 8; ++s) {
            const int ka = 4 * s + 2 * half;          // K index held in VGPR0
            // unconditional 8B-aligned LDS load -> ds_load_b64
            v2f a = *(const v2f*)&s_dpi[wave][lm * DD + ka];
            v2f b0, b1;                                // cov rows ka, ka+1
            b0[0] = cv[ka       * DD + lm];
            b0[1] = cv[(ka + 1) * DD + lm];
            b1[0] = cv[ka       * DD + 16 + lm];
            b1[1] = cv[(ka + 1) * DD + 16 + lm];
            accT0 = __builtin_amdgcn_wmma_f32_16x16x4_f32(
                        false, a, false, b0, (short)0, accT0, false, false);
            accT1 = __builtin_amdgcn_wmma_f32_16x16x4_f32(
                        false, a, false, b1, (short)0, accT1, false, false);
        }
        // spill T to LDS in row-major so it can be re-read in A layout
        #pragma unroll
        for (int v = 0; v < 8; ++v) {
            const int m = v + 8 * half;
            s_T[wave][m * DD + lm]      = accT0[v];
            s_T[wave][m * DD + 16 + lm] = accT1[v];
        }
    }
    __syncthreads();

    // ---- Phase C: bbt(8x8 in 16x16) = T @ pad(dpi)^T -----------------------
    {
        v8f accB = {};
        #pragma unroll
        for (int s = 0; s < 8; ++s) {
            const int ka = 4 * s + 2 * half;
            // A from T, B'[k][m] = dpi[m][k] (cols m>=8 read zero padding)
            v2f a = *(const v2f*)&s_T  [wave][lm * DD + ka];
            v2f b = *(const v2f*)&s_dpi[wave][lm * DD + ka];
            accB = __builtin_amdgcn_wmma_f32_16x16x4_f32(
                        false, a, false, b, (short)0, accB, false, false);
        }
        // D layout: VGPR v / half 0 -> row v; only rows<8, cols<8 meaningful
        if (lane < dd) {
            #pragma unroll
            for (int v = 0; v < 8; ++v)
                s_bbt[wave][v * dd + lane] = accB[v];
        }
    }
    __syncthreads();

    // ---- Phase D: qv[i] = sum_{j,k} bbt[j,k] * Hd[i,k,j]; t; mse -----------
    {
        const v4f* hd = (const v4f*)(dh + (size_t)n * (DD * dd * dd)
                                        + (size_t)lane * (dd * dd));
        float qv = 0.f;
        #pragma unroll
        for (int k = 0; k < dd; ++k) {
            v4f h0 = hd[k * 2 + 0];   // Hd[lane][k][0..3]
            v4f h1 = hd[k * 2 + 1];   // Hd[lane][k][4..7]
            qv += s_bbt[wave][0 * dd + k] * h0.x + s_bbt[wave][1 * dd + k] * h0.y
                + s_bbt[wave][2 * dd + k] * h0.z + s_bbt[wave][3 * dd + k] * h0.w
                + s_bbt[wave][4 * dd + k] * h1.x + s_bbt[wave][5 * dd + k] * h1.y
                + s_bbt[wave][6 * dd + k] * h1.z + s_bbt[wave][7 * dd + k] * h1.w;
        }
        const float tv = drift[(size_t)n * DD + lane] - 0.5f * qv;
        s_t[wave][lane] = tv;
        const float dx = x_hat[(size_t)n * DD + lane] - x[(size_t)n * DD + lane];
        sum_mse += dx * dx;
    }
    __syncthreads();

    // ---- Phase E: npv = t - P@t (row per lane) + tangent diff^2 ------------
    {
        const float* oprow = op + (size_t)n * (DD * DD) + (size_t)lane * DD;
        const float* mprow = mp + (size_t)n * (DD * DD) + (size_t)lane * DD;
        float acc = 0.f;
        #pragma unroll
        for (int j = 0; j < DD; ++j) {
            const float o = oprow[j];
            const float m = mprow[j];
            acc += o * s_t[wave][j];
            const float dmo = m - o;
            sum_tang += dmo * dmo;
        }
        const float npv = s_t[wave][lane] - acc;
        sum_curv += npv * npv;
    }

    // ---- Phase F: encoder_hessian sum of squares (655 MB stream, NT b128) --
    {
        const v4f* ehp = (const v4f*)(eh + (size_t)n * (dd * DD * DD));
        #pragma unroll 4
        for (int k = 0; k < 64; ++k) {
            v4f v = __builtin_nontemporal_load(&ehp[k * 32 + lane]);
            sum_hess += v.x * v.x + v.y * v.y + v.z * v.z + v.w * v.w;
        }
    }

    // ---- combine with reference weights ------------------------------------
    float total = sum_mse * (1.0f / (float)(NN * DD))
                + (sum_contr + sum_hess + sum_tang) * (1.0f / (float)NN)
                + sum_curv;

    // wave32 reduction
    #pragma unroll
    for (int off = 16; off > 0; off >>= 1)
        total += __shfl_xor(total, off, 32);
    if (lane == 0) s_part[wave] = total;
    __syncthreads();
    if (threadIdx.x == 0) {
        float b = 0.f;
        #pragma unroll
        for (int w = 0; w < WPB; ++w) b += s_part[w];
        partial[blockIdx.x] = b;
    }
}

// ---------------------------------------------------------------------------
// Deterministic final reduction of 2500 block partials -> d_out[0]
// ---------------------------------------------------------------------------
__global__ __launch_bounds__(256) void cc2_reduce(const float* __restrict__ partial,
                                                  float* __restrict__ out,
                                                  int nblocks)
{
    __shared__ float sh[256];
    float s = 0.f;
    for (int i = threadIdx.x; i < nblocks; i += 256) s += partial[i];
    sh[threadIdx.x] = s;
    __syncthreads();
    for (int stride = 128; stride > 0; stride >>= 1) {
        if (threadIdx.x < stride) sh[threadIdx.x] += sh[threadIdx.x + stride];
        __syncthreads();
    }
    if (threadIdx.x == 0) out[0] = sh[0];
}

extern "C" void kernel_launch(void* const* d_in, const int* in_sizes, int n_in,
                              void* d_out, int out_size, void* d_ws, size_t ws_size,
                              hipStream_t stream)
{
    const float* x_hat = (const float*)d_in[0];
    const float* dpi   = (const float*)d_in[1];
    const float* mp    = (const float*)d_in[2];
    const float* dh    = (const float*)d_in[3];
    const float* eh    = (const float*)d_in[4];
    const float* x     = (const float*)d_in[5];
    const float* drift = (const float*)d_in[6];
    const float* cov   = (const float*)d_in[7];
    const float* op    = (const float*)d_in[8];

    float* partial = (float*)d_ws;   // NBLK floats = 10 KB scratch

    cc2_loss_main<<<NBLK, BLK, 0, stream>>>(x_hat, dpi, mp, dh, eh, x, drift,
                                            cov, op, partial);
    cc2_reduce<<<1, 256, 0, stream>>>(partial, (float*)d_out, NBLK);
}